// BasicBlock_58420145160619
// MI455X (gfx1250) — compile-verified
//
#include <hip/hip_runtime.h>

#define N_VOX 200000
#define CCH   64
#define NPAIR 65536
#define BN_EPS 1e-5f

typedef float v2f __attribute__((ext_vector_type(2)));
typedef float v8f __attribute__((ext_vector_type(8)));

__device__ __forceinline__ v8f wmma_f32(v2f a, v2f b, v8f c) {
  // D = A(16x4) * B(4x16) + C, exact fp32 (RNE)
  return __builtin_amdgcn_wmma_f32_16x16x4_f32(false, a, false, b, (short)0, c,
                                               false, false);
}

// Cooperative fill of LDS with W (64x64) interleaved by K-pairs:
//   sW2[(K>>1)*128 + n*2 + (K&1)] = W[K][n]
// so that lane operand (W[K][n], W[K+1][n]) is one aligned 8-byte word
// -> single ds_load_b64 per B operand, no VGPR repacking.
__device__ __forceinline__ void fill_w_interleaved(float* sW2,
                                                   const float* __restrict__ W,
                                                   int tid) {
  for (int i = tid; i < CCH * CCH; i += 256) {
    const int K = i >> 6, n = i & 63;
    sW2[(K >> 1) * 128 + n * 2 + (K & 1)] = W[i];
  }
}

// Per-wave 16x64 GEMM against LDS-interleaved W. A operand supplied by caller
// via base pointer (row already resolved); `ok` zero-fills gathered rows that
// hit the sentinel. Produces 4 accumulators (N-tiles of 16).
__device__ __forceinline__ void wave_gemm_16x64(const float* arow, bool ok,
                                                const float* sW2, int lm,
                                                int hi, v8f& acc0, v8f& acc1,
                                                v8f& acc2, v8f& acc3) {
#pragma unroll
  for (int kk = 0; kk < 16; ++kk) {
    v2f a;
    a.x = ok ? arow[4 * kk] : 0.f;
    a.y = ok ? arow[4 * kk + 1] : 0.f;
    const v2f* bp = (const v2f*)(sW2 + (2 * kk + hi) * 128 + 2 * lm);
    v2f b0 = bp[0];   // n0 = 0
    v2f b1 = bp[16];  // n0 = 16
    v2f b2 = bp[32];  // n0 = 32
    v2f b3 = bp[48];  // n0 = 48
    acc0 = wmma_f32(a, b0, acc0);
    acc1 = wmma_f32(a, b1, acc1);
    acc2 = wmma_f32(a, b2, acc2);
    acc3 = wmma_f32(a, b3, acc3);
  }
}

// ---------------------------------------------------------------------------
// Center GEMM: hdst[0:N,:] = xsrc[0:N,:] @ Wc(64x64). Also zeroes dump row
// hdst[N,:] and (optionally) the 256-float BN stat accumulators.
// ---------------------------------------------------------------------------
__global__ void __launch_bounds__(256) conv_center(
    const float* __restrict__ xsrc, const float* __restrict__ Wc,
    float* __restrict__ hdst, float* __restrict__ stats_zero, int ntiles) {
  __shared__ float sW2[CCH * CCH];
  const int tid = threadIdx.x;
  fill_w_interleaved(sW2, Wc, tid);

  if (blockIdx.x == 0) {
    if (stats_zero) stats_zero[tid] = 0.f;              // 256 stat floats
    if (tid < CCH) hdst[(long)N_VOX * CCH + tid] = 0.f; // dump row
  }
  __syncthreads();

  const int wave = tid >> 5, lane = tid & 31;
  const int lm = lane & 15, hi = lane >> 4;
  const int tile = blockIdx.x * 8 + wave;
  if (tile >= ntiles) return;
  const int m0 = tile * 16;

  v8f acc0 = {}, acc1 = {}, acc2 = {}, acc3 = {};
  const float* arow = xsrc + (long)(m0 + lm) * CCH + 2 * hi;
  wave_gemm_16x64(arow, true, sW2, lm, hi, acc0, acc1, acc2, acc3);

  float* drow = hdst + (long)(m0 + 8 * hi) * CCH + lm;
#pragma unroll
  for (int v = 0; v < 8; ++v) {
    drow[v * CCH + 0]  = acc0[v];
    drow[v * CCH + 16] = acc1[v];
    drow[v * CCH + 32] = acc2[v];
    drow[v * CCH + 48] = acc3[v];
  }
}

// ---------------------------------------------------------------------------
// Offset conv: for offset k, tile of 16 pairs: gather rows via in_maps,
// GEMM with W_off[k] (LDS-interleaved), scatter-add via out_maps (f32 atomics)
// grid = 26*512 blocks; each block fully inside one offset k.
// ---------------------------------------------------------------------------
__global__ void __launch_bounds__(256) conv_scatter(
    const float* __restrict__ xsrc, const float* __restrict__ W,
    const int* __restrict__ in_maps, const int* __restrict__ out_maps,
    float* __restrict__ hdst, int nrows_src) {
  __shared__ float sW2[CCH * CCH];
  const int tid = threadIdx.x;
  const int k = blockIdx.x >> 9;    // offset index 0..25
  const int tb = blockIdx.x & 511;  // tile-block within offset
  const int wk = (k < 13) ? k : (k + 1);
  fill_w_interleaved(sW2, W + (long)wk * CCH * CCH, tid);
  __syncthreads();

  const int wave = tid >> 5, lane = tid & 31;
  const int lm = lane & 15, hi = lane >> 4;
  const int p0 = (tb * 8 + wave) * 16;
  const int* im = in_maps + (long)k * NPAIR + p0;
  const int* om = out_maps + (long)k * NPAIR + p0;

  const int row = im[lm];
  const bool ok = row < nrows_src;  // sentinel N -> zero row
  const float* arow = xsrc + (long)row * CCH + 2 * hi;

  v8f acc0 = {}, acc1 = {}, acc2 = {}, acc3 = {};
  wave_gemm_16x64(arow, ok, sW2, lm, hi, acc0, acc1, acc2, acc3);

#pragma unroll
  for (int v = 0; v < 8; ++v) {
    const int r = om[v + 8 * hi];   // sentinel N -> dump row (exists)
    float* drow = hdst + (long)r * CCH + lm;
    atomicAdd(drow + 0,  acc0[v]);
    atomicAdd(drow + 16, acc1[v]);
    atomicAdd(drow + 32, acc2[v]);
    atomicAdd(drow + 48, acc3[v]);
  }
}

// ---------------------------------------------------------------------------
// Per-channel sum / sum-of-squares over h[0:N,:]  (two-level reduction)
// ---------------------------------------------------------------------------
__global__ void __launch_bounds__(256) bn_stats(const float* __restrict__ h,
                                                float* __restrict__ ssum,
                                                float* __restrict__ ssq) {
  __shared__ float redS[256], redQ[256];
  const int tid = threadIdx.x;
  const int c = tid & 63;
  int r = (blockIdx.x * 256 + tid) >> 6;
  const int stride = (gridDim.x * 256) >> 6;
  float s = 0.f, q = 0.f;
  for (; r < N_VOX; r += stride) {
    float v = h[(long)r * CCH + c];
    s += v;
    q += v * v;
  }
  redS[tid] = s;
  redQ[tid] = q;
  __syncthreads();
  if (tid < 64) {
    s = redS[tid] + redS[tid + 64] + redS[tid + 128] + redS[tid + 192];
    q = redQ[tid] + redQ[tid + 64] + redQ[tid + 128] + redQ[tid + 192];
    atomicAdd(&ssum[c], s);
    atomicAdd(&ssq[c], q);
  }
}

// y = relu(gamma*(h-mu)*rsqrt(var+eps)+beta); dump row -> 0
__global__ void __launch_bounds__(256) bn_relu(
    const float* __restrict__ h, const float* __restrict__ ssum,
    const float* __restrict__ ssq, const float* __restrict__ gamma,
    const float* __restrict__ beta, float* __restrict__ y, long total) {
  long i = (long)blockIdx.x * 256 + threadIdx.x;
  if (i >= total) return;
  const int c = (int)(i & 63);
  if ((i >> 6) >= N_VOX) { y[i] = 0.f; return; }
  const float inv_n = 1.f / (float)N_VOX;
  const float mu = ssum[c] * inv_n;
  const float var = ssq[c] * inv_n - mu * mu;
  const float a = gamma[c] * rsqrtf(var + BN_EPS);
  const float b = beta[c] - mu * a;
  y[i] = fmaxf(a * h[i] + b, 0.f);
}

// out = relu(bn(h2) + x)
__global__ void __launch_bounds__(256) bn_add_relu(
    const float* __restrict__ h, const float* __restrict__ x,
    const float* __restrict__ ssum, const float* __restrict__ ssq,
    const float* __restrict__ gamma, const float* __restrict__ beta,
    float* __restrict__ out, long total) {
  long i = (long)blockIdx.x * 256 + threadIdx.x;
  if (i >= total) return;
  const int c = (int)(i & 63);
  const float inv_n = 1.f / (float)N_VOX;
  const float mu = ssum[c] * inv_n;
  const float var = ssq[c] * inv_n - mu * mu;
  const float a = gamma[c] * rsqrtf(var + BN_EPS);
  const float b = beta[c] - mu * a;
  out[i] = fmaxf(a * h[i] + b + x[i], 0.f);
}

extern "C" void kernel_launch(void* const* d_in, const int* in_sizes, int n_in,
                              void* d_out, int out_size, void* d_ws,
                              size_t ws_size, hipStream_t stream) {
  const float* x  = (const float*)d_in[0];
  const float* W1 = (const float*)d_in[1];
  const float* g1 = (const float*)d_in[2];
  const float* b1 = (const float*)d_in[3];
  const float* W2 = (const float*)d_in[4];
  const float* g2 = (const float*)d_in[5];
  const float* b2 = (const float*)d_in[6];
  const int* im   = (const int*)d_in[7];
  const int* om   = (const int*)d_in[8];
  float* out = (float*)d_out;

  float* ws = (float*)d_ws;
  const long rowsz = (long)(N_VOX + 1) * CCH;  // 12,800,064 floats
  float* bufH = ws;                 // h (conv1), then h2 (conv2)
  float* bufY = ws + rowsz;         // relu(bn(h))
  float* stats = ws + 2 * rowsz;    // 256 floats
  float* s1sum = stats, *s1sq = stats + 64;
  float* s2sum = stats + 128, *s2sq = stats + 192;

  const int ntiles = N_VOX / 16;            // 12500
  const int gC = (ntiles + 7) / 8;          // 1563 blocks (8 waves each)
  const int gS = 26 * 512;                  // one offset per 512 blocks
  dim3 blk(256);

  // ---- conv1: h = sparse_conv(x, W1) ----
  conv_center<<<gC, blk, 0, stream>>>(x, W1 + 13 * CCH * CCH, bufH, stats,
                                      ntiles);
  conv_scatter<<<gS, blk, 0, stream>>>(x, W1, im, om, bufH, N_VOX);
  bn_stats<<<1024, blk, 0, stream>>>(bufH, s1sum, s1sq);
  bn_relu<<<(int)((rowsz + 255) / 256), blk, 0, stream>>>(bufH, s1sum, s1sq,
                                                          g1, b1, bufY, rowsz);

  // ---- conv2: h2 = sparse_conv(y, W2)  (h2 reuses bufH) ----
  conv_center<<<gC, blk, 0, stream>>>(bufY, W2 + 13 * CCH * CCH, bufH, nullptr,
                                      ntiles);
  conv_scatter<<<gS, blk, 0, stream>>>(bufY, W2, im, om, bufH, N_VOX + 1);
  bn_stats<<<1024, blk, 0, stream>>>(bufH, s2sum, s2sq);

  const long totO = (long)N_VOX * CCH;
  bn_add_relu<<<(int)((totO + 255) / 256), blk, 0, stream>>>(
      bufH, x, s2sum, s2sq, g2, b2, out, totO);
}